// MambaSimple_24524263260374
// MI455X (gfx1250) — compile-verified
//
#include <hip/hip_runtime.h>
#include <hip/hip_bf16.h>
#include <math.h>

typedef __attribute__((ext_vector_type(2))) float v2f;
typedef __attribute__((ext_vector_type(8))) float v8f;

#define D_MODEL 2048
#define D_STATE 16
#define D_CONV  4
#define DT_RANK 128
#define D_INNER 4096
#define BATCH   4
#define SEQLEN  2048
#define MROWS   (BATCH * SEQLEN)   // 8192 token rows

// ---------------- fp32 WMMA GEMM:  C[M,N] = A[M,K] * W[N,K]^T (+ optional epilogue)
// Block tile 64x128, 256 threads = 8 waves (2x4), wave tile 32x32 = 2x2 WMMA frags.
// K staged through LDS in chunks of 16 (4 x K=4 WMMA steps).
// ASY=1: double-buffered GLOBAL_LOAD_ASYNC_TO_LDS_B128 pipeline (full tiles only).
// ASY=0: guarded synchronous staging (partial-N tiles) + global_prefetch of next chunk.
#define BLK_M 64
#define BLK_N 128
#define KC    16
#define LDSP  20   // padded LDS row (floats): 16B aligned, bank-conflict free

// Issue one 16-byte async global->LDS copy (per-lane addresses, ASYNCcnt-tracked).
__device__ __forceinline__ void async_copy_b128(const float* gsrc, const float* ldst)
{
    const unsigned long long ga = (unsigned long long)(uintptr_t)gsrc;
    const unsigned           la = (unsigned)(uintptr_t)ldst;   // low 32b of flat = LDS offset
    asm volatile("global_load_async_to_lds_b128 %0, %1, off"
                 :: "v"(la), "v"(ga) : "memory");
}

template<int EPI, int ASY>
__global__ __launch_bounds__(256)
void wmma_gemm_f32(const float* __restrict__ A, const float* __restrict__ W,
                   float* __restrict__ C, const float* __restrict__ bias,
                   int M, int N, int K, int lda, int ldw, int ldc)
{
    __shared__ float As[2][BLK_M * LDSP];   // [buf][row][k]  activations
    __shared__ float Bs[2][BLK_N * LDSP];   // [buf][n][k]    weight rows (W is N x K)

    const int tid  = threadIdx.x;
    const int lane = tid & 31;
    const int wave = tid >> 5;           // 0..7
    const int wm   = wave >> 2;          // 0..1 (M)
    const int wn   = wave & 3;           // 0..3 (N)
    const int r    = lane & 15;
    const int half = lane >> 4;

    const int blockN = blockIdx.x * BLK_N;
    const int blockM = blockIdx.y * BLK_M;

    // per-thread staging coordinates
    const int arow = tid >> 2;           // 0..63
    const int akq  = (tid & 3) * 4;      // 0/4/8/12
    const int bn0  = tid >> 2;           // 0..63
    const int bn1  = bn0 + 64;           // 64..127
    const int bkq  = (tid & 3) * 4;

    v8f acc[2][2] = {};

    // fragment compute on LDS buffer `buf`
    auto compute = [&](int buf) {
        #pragma unroll
        for (int ks = 0; ks < 4; ++ks) {
            const int kk = ks * 4 + 2 * half;   // lanes 0-15: k=kk,kk+1 ; 16-31: +2
            v2f a[2], b[2];
            #pragma unroll
            for (int mt = 0; mt < 2; ++mt) {
                const int row = wm * 32 + mt * 16 + r;
                a[mt].x = As[buf][row * LDSP + kk];
                a[mt].y = As[buf][row * LDSP + kk + 1];
            }
            #pragma unroll
            for (int nt = 0; nt < 2; ++nt) {
                const int col = wn * 32 + nt * 16 + r;
                b[nt].x = Bs[buf][col * LDSP + kk];
                b[nt].y = Bs[buf][col * LDSP + kk + 1];
            }
            #pragma unroll
            for (int mt = 0; mt < 2; ++mt)
                #pragma unroll
                for (int nt = 0; nt < 2; ++nt)
                    acc[mt][nt] = __builtin_amdgcn_wmma_f32_16x16x4_f32(
                        false, a[mt], false, b[nt],
                        (short)0, acc[mt][nt], false, false);
        }
    };

    if (ASY) {
        // ---- async double-buffered pipeline (requires full N tile)
        auto issue = [&](int kb, int buf) {
            async_copy_b128(A + (size_t)(blockM + arow) * lda + kb + akq,
                            &As[buf][arow * LDSP + akq]);
            async_copy_b128(W + (size_t)(blockN + bn0) * ldw + kb + bkq,
                            &Bs[buf][bn0 * LDSP + bkq]);
            async_copy_b128(W + (size_t)(blockN + bn1) * ldw + kb + bkq,
                            &Bs[buf][bn1 * LDSP + bkq]);
        };
        const int nch = K / KC;
        issue(0, 0);
        for (int c = 0; c < nch; ++c) {
            const int buf = c & 1;
            if (c + 1 < nch) {
                issue((c + 1) * KC, buf ^ 1);                     // prefetch next chunk
                asm volatile("s_wait_asynccnt 0x3" ::: "memory"); // chunk c landed
            } else {
                asm volatile("s_wait_asynccnt 0x0" ::: "memory");
            }
            __syncthreads();          // all waves' async writes visible
            compute(buf);
            __syncthreads();          // buffer reuse fence (chunk c+2 overwrites buf)
        }
    } else {
        // ---- synchronous guarded path (partial N), with next-chunk prefetch hints
        for (int kb = 0; kb < K; kb += KC) {
            if (kb + KC < K) {
                __builtin_prefetch(A + (size_t)(blockM + arow) * lda + kb + KC + akq, 0, 0);
                if (blockN + bn0 < N)
                    __builtin_prefetch(W + (size_t)(blockN + bn0) * ldw + kb + KC + bkq, 0, 0);
            }
            {
                const float4 av = *reinterpret_cast<const float4*>(
                    A + (size_t)(blockM + arow) * lda + kb + akq);
                *reinterpret_cast<float4*>(&As[0][arow * LDSP + akq]) = av;
            }
            #pragma unroll
            for (int i = 0; i < 2; ++i) {
                const int n  = (i == 0) ? bn0 : bn1;
                const int gn = blockN + n;
                float4 wv = make_float4(0.f, 0.f, 0.f, 0.f);
                if (gn < N)
                    wv = *reinterpret_cast<const float4*>(W + (size_t)gn * ldw + kb + bkq);
                *reinterpret_cast<float4*>(&Bs[0][n * LDSP + bkq]) = wv;
            }
            __syncthreads();
            compute(0);
            __syncthreads();
        }
    }

    // --- epilogue + store (coalesced across lanes; ldc-strided per element)
    #pragma unroll
    for (int mt = 0; mt < 2; ++mt) {
        #pragma unroll
        for (int nt = 0; nt < 2; ++nt) {
            const int col = blockN + wn * 32 + nt * 16 + r;
            if (col >= N) continue;
            #pragma unroll
            for (int i = 0; i < 8; ++i) {
                const int row = blockM + wm * 32 + mt * 16 + half * 8 + i;
                float v = acc[mt][nt][i];
                if (EPI == 1) {                       // softplus(v + dt_bias[col])
                    v += bias[col];
                    v = (v > 20.0f) ? v : log1pf(__expf(v));
                }
                C[(size_t)row * ldc + col] = v;
            }
        }
    }
}

// ---------------- depthwise causal conv1d (k=4) + bias + SiLU, in place on x-half of xz.
// One thread per (b, d) channel; rolling register window makes in-place safe.
__global__ __launch_bounds__(256)
void conv_silu_kernel(float* __restrict__ xz,
                      const float* __restrict__ cw, const float* __restrict__ cb)
{
    const int t = blockIdx.x * 256 + threadIdx.x;     // 16384 threads
    const int b = t >> 12;
    const int d = t & (D_INNER - 1);
    const float w0 = cw[d * 4 + 0], w1 = cw[d * 4 + 1];
    const float w2 = cw[d * 4 + 2], w3 = cw[d * 4 + 3];
    const float bias = cb[d];
    float x0 = 0.f, x1 = 0.f, x2 = 0.f;
    float* p = xz + (size_t)b * SEQLEN * (2 * D_INNER) + d;
    for (int l = 0; l < SEQLEN; ++l) {
        const float x3 = p[(size_t)l * (2 * D_INNER)];
        float v = w0 * x0 + w1 * x1 + w2 * x2 + w3 * x3 + bias;
        v = v / (1.0f + __expf(-v));                  // SiLU
        p[(size_t)l * (2 * D_INNER)] = v;
        x0 = x1; x1 = x2; x2 = x3;
    }
}

// ---------------- selective scan, one thread per (b, d); h[16] in registers.
// Reads delta from `dy`, writes gated y back in place (read-before-write per element).
__global__ __launch_bounds__(256)
void scan_kernel(const float* __restrict__ xz, const float* __restrict__ xdbl,
                 float* __restrict__ dy,
                 const float* __restrict__ A_log, const float* __restrict__ Dv)
{
    const int b = blockIdx.x >> 4;                    // 16 blocks per batch
    const int d = ((blockIdx.x & 15) << 8) + threadIdx.x;

    float Arow[D_STATE];
    #pragma unroll
    for (int n = 0; n < D_STATE; ++n)
        Arow[n] = -__expf(A_log[d * D_STATE + n]);
    const float Dd = Dv[d];

    float h[D_STATE];
    #pragma unroll
    for (int n = 0; n < D_STATE; ++n) h[n] = 0.f;

    const float* px  = xz + (size_t)b * SEQLEN * (2 * D_INNER) + d;          // xact
    const float* pz  = px + D_INNER;                                          // z
    float*       pdy = dy + (size_t)b * SEQLEN * D_INNER + d;                 // delta -> y
    const float* pbc = xdbl + (size_t)b * SEQLEN * (DT_RANK + 2 * D_STATE);

    for (int l = 0; l < SEQLEN; ++l) {
        const float u   = px[(size_t)l * (2 * D_INNER)];
        const float zz  = pz[(size_t)l * (2 * D_INNER)];
        const float dlt = pdy[(size_t)l * D_INNER];
        const float* bc = pbc + (size_t)l * (DT_RANK + 2 * D_STATE);
        const float du  = dlt * u;
        float y = 0.f;
        #pragma unroll
        for (int n = 0; n < D_STATE; ++n) {
            const float dA = __expf(dlt * Arow[n]);
            h[n] = h[n] * dA + du * bc[DT_RANK + n];          // B broadcast load
            y += h[n] * bc[DT_RANK + D_STATE + n];            // C broadcast load
        }
        y += Dd * u;
        const float sg = zz / (1.0f + __expf(-zz));           // SiLU gate
        pdy[(size_t)l * D_INNER] = y * sg;
    }
}

extern "C" void kernel_launch(void* const* d_in, const int* in_sizes, int n_in,
                              void* d_out, int out_size, void* d_ws, size_t ws_size,
                              hipStream_t stream)
{
    const float* hs   = (const float*)d_in[0];   // (B,L,2048)
    const float* Win  = (const float*)d_in[1];   // (8192,2048)
    const float* cw   = (const float*)d_in[2];   // (4096,4)
    const float* cb   = (const float*)d_in[3];   // (4096,)
    const float* Wx   = (const float*)d_in[4];   // (160,4096)
    const float* Wdt  = (const float*)d_in[5];   // (4096,128)
    const float* dtb  = (const float*)d_in[6];   // (4096,)
    const float* Alog = (const float*)d_in[7];   // (4096,16)
    const float* Dv   = (const float*)d_in[8];   // (4096,)
    const float* Wout = (const float*)d_in[9];   // (2048,4096)
    float* out = (float*)d_out;

    char* ws = (char*)d_ws;
    float* xz    = (float*)(ws);                                   // 8192 x 8192  (256 MB)
    float* xdbl  = (float*)(ws + 268435456UL);                     // 8192 x 160   (5.25 MB)
    float* delta = (float*)(ws + 268435456UL + 5242880UL);         // 8192 x 4096  (128 MB)

    const dim3 blk(256);

    // 1) in_proj: xz = hidden @ W_in^T            M=8192 N=8192 K=2048  (async pipeline)
    wmma_gemm_f32<0, 1><<<dim3(8192 / BLK_N, MROWS / BLK_M), blk, 0, stream>>>(
        hs, Win, xz, nullptr, MROWS, 2 * D_INNER, D_MODEL, D_MODEL, D_MODEL, 2 * D_INNER);

    // 2) causal depthwise conv + SiLU, in place on x-half of xz
    conv_silu_kernel<<<dim3(64), blk, 0, stream>>>(xz, cw, cb);

    // 3) x_proj: x_dbl = xact @ W_x^T             M=8192 N=160 K=4096 (guarded sync path)
    wmma_gemm_f32<0, 0><<<dim3((160 + BLK_N - 1) / BLK_N, MROWS / BLK_M), blk, 0, stream>>>(
        xz, Wx, xdbl, nullptr, MROWS, DT_RANK + 2 * D_STATE, D_INNER,
        2 * D_INNER, D_INNER, DT_RANK + 2 * D_STATE);

    // 4) dt: delta = softplus(dt_lr @ W_dt^T + dt_bias)   M=8192 N=4096 K=128  (async)
    wmma_gemm_f32<1, 1><<<dim3(D_INNER / BLK_N, MROWS / BLK_M), blk, 0, stream>>>(
        xdbl, Wdt, delta, dtb, MROWS, D_INNER, DT_RANK,
        DT_RANK + 2 * D_STATE, DT_RANK, D_INNER);

    // 5) selective scan + D-skip + SiLU(z) gate; y overwrites delta
    scan_kernel<<<dim3(64), blk, 0, stream>>>(xz, xdbl, delta, Alog, Dv);

    // 6) out_proj: out = y @ W_out^T              M=8192 N=2048 K=4096  (async)
    wmma_gemm_f32<0, 1><<<dim3(D_MODEL / BLK_N, MROWS / BLK_M), blk, 0, stream>>>(
        delta, Wout, out, nullptr, MROWS, D_MODEL, D_INNER, D_INNER, D_INNER, D_MODEL);
}